// InterfaceGroupModule_43800076485390
// MI455X (gfx1250) — compile-verified
//
#include <hip/hip_runtime.h>

typedef __attribute__((ext_vector_type(2))) float    v2f;
typedef __attribute__((ext_vector_type(8))) float    v8f;
typedef __attribute__((ext_vector_type(4))) unsigned u32x4;
typedef __attribute__((ext_vector_type(8))) int      i32x8;
typedef __attribute__((ext_vector_type(4))) int      i32x4;

#define B_    16
#define N_    8192
#define HALF_ 4096
#define S_    512      // FPS samples per half
#define G_    1024     // total groups per batch
#define K_    32
#define C_    64

// ---------------------------------------------------------------------------
// Tensor Data Mover: 1-D contiguous DWORD copy global -> LDS.
// D# layout per CDNA5 ISA (count=1, type=2,
// data_size=4B, tensor_dim0=tile_dim0=nelem dwords, tensor_dim1=1).
// This toolchain exposes the 6-arg builtin (groups 2/3 + trailing group all
// zero for a <=2-D tensor). Issue from ONE wave; caller must
// s_wait_tensorcnt + barrier.
// ---------------------------------------------------------------------------
__device__ __forceinline__ void tdm_load_1d(unsigned lds_off, const void* gaddr,
                                            unsigned nelem) {
  unsigned long long ga = (unsigned long long)gaddr;
  unsigned ga_lo = (unsigned)__builtin_amdgcn_readfirstlane((int)(unsigned)ga);
  unsigned ga_hi = (unsigned)__builtin_amdgcn_readfirstlane((int)(unsigned)(ga >> 32));
  unsigned ldsa  = (unsigned)__builtin_amdgcn_readfirstlane((int)lds_off);

  u32x4 g0;
  g0[0] = 1u;                                      // count=1, user mode
  g0[1] = ldsa;                                    // lds_addr (bytes)
  g0[2] = ga_lo;                                   // global_addr[31:0]
  g0[3] = (ga_hi & 0x01FFFFFFu) | 0x80000000u;     // global_addr[56:32] | type=2

  i32x8 g1;
  g1[0] = 0x00020000;                              // workgroup_mask=0, data_size=4B
  g1[1] = (int)((nelem & 0xFFFFu) << 16);          // tensor_dim0[15:0]  @ bits 63:48
  g1[2] = (int)((nelem >> 16) | (1u << 16));       // tensor_dim0[31:16], tensor_dim1=1
  g1[3] = (int)((nelem & 0xFFFFu) << 16);          // tile_dim0 = nelem  @ bits 127:112
  g1[4] = 0;                                       // tile_dim1=0, tile_dim2=0
  g1[5] = (int)nelem;                              // tensor_dim0_stride lo
  g1[6] = 0;
  g1[7] = 0;

  i32x4 z4 = {0, 0, 0, 0};
  i32x8 z8 = {0, 0, 0, 0, 0, 0, 0, 0};
  __builtin_amdgcn_tensor_load_to_lds(g0, g1, z4, z4, z8, 0);
}

// ---------------------------------------------------------------------------
// Block-wide argmax with first-index tie break (matches jnp.argmax).
// ---------------------------------------------------------------------------
__device__ __forceinline__ int block_argmax(float v, int i, float* swv, int* swi, int* s_out) {
  const int lane = threadIdx.x & 31;
  const int wave = threadIdx.x >> 5;
  #pragma unroll
  for (int off = 16; off > 0; off >>= 1) {
    float ov = __shfl_down(v, off, 32);
    int   oi = __shfl_down(i, off, 32);
    if (ov > v || (ov == v && oi < i)) { v = ov; i = oi; }
  }
  if (lane == 0) { swv[wave] = v; swi[wave] = i; }
  __syncthreads();
  if (threadIdx.x == 0) {
    float bv = swv[0]; int bi = swi[0];
    #pragma unroll
    for (int w = 1; w < 8; ++w)
      if (swv[w] > bv || (swv[w] == bv && swi[w] < bi)) { bv = swv[w]; bi = swi[w]; }
    *s_out = bi;
  }
  __syncthreads();
  return *s_out;
}

// ---------------------------------------------------------------------------
// Farthest point sampling: one block per (batch, half). 256 threads, 16 pts ea.
// xyz half staged into LDS by the Tensor Data Mover (48KB, one descriptor).
// ---------------------------------------------------------------------------
__global__ __launch_bounds__(256) void fps_kernel(const float* __restrict__ xyz,
                                                  int* __restrict__ sel) {
  extern __shared__ float sx[];   // HALF_*3 floats (48KB dynamic)
  __shared__ float swv[8];
  __shared__ int   swi[8];
  __shared__ int   s_cur;

  const int tid  = threadIdx.x;
  const int b    = blockIdx.x >> 1;
  const int half = blockIdx.x & 1;
  const int base = (b * N_ + half * HALF_) * 3;

  if (tid < 32) {
    tdm_load_1d((unsigned)(size_t)&sx[0], xyz + base, HALF_ * 3);
    __builtin_amdgcn_s_wait_tensorcnt(0);
  }
  __syncthreads();

  float md[16];
  float bv = -1e30f; int bi = 0;
  #pragma unroll
  for (int j = 0; j < 16; ++j) {
    const int p = j*256 + tid;
    const float x = sx[p*3], y = sx[p*3+1], z = sx[p*3+2];
    const float n2 = x*x + y*y + z*z;
    md[j] = 1e10f;
    if (n2 > bv) { bv = n2; bi = p; }
  }
  int cur = block_argmax(bv, bi, swv, swi, &s_cur);

  for (int it = 0; it < S_; ++it) {
    if (tid == 0) sel[blockIdx.x * S_ + it] = cur;
    const float cx = sx[cur*3], cy = sx[cur*3+1], cz = sx[cur*3+2];
    bv = -1e30f; bi = 0;
    #pragma unroll
    for (int j = 0; j < 16; ++j) {
      const int p = j*256 + tid;
      const float dx = sx[p*3]-cx, dy = sx[p*3+1]-cy, dz = sx[p*3+2]-cz;
      const float d = dx*dx + dy*dy + dz*dz;
      md[j] = fminf(md[j], d);
      if (md[j] > bv) { bv = md[j]; bi = p; }
    }
    cur = block_argmax(bv, bi, swv, swi, &s_cur);
  }
}

// ---------------------------------------------------------------------------
// Gather lc_xyz / lc_feat. One block per (b, g), 64 threads.
// ---------------------------------------------------------------------------
__global__ __launch_bounds__(64) void lc_gather_kernel(const float* __restrict__ xyz,
                                                       const float* __restrict__ feat,
                                                       const int* __restrict__ sel,
                                                       float* __restrict__ lc_xyz,
                                                       float* __restrict__ lc_feat) {
  const int bg  = blockIdx.x;
  const int b   = bg >> 10;
  const int g   = bg & 1023;
  const int half = g >> 9;
  const int idx = sel[(b*2 + half)*S_ + (g & 511)];
  const int row = b*N_ + half*HALF_ + idx;
  const int t = threadIdx.x;
  lc_feat[(b*G_ + g)*C_ + t] = feat[row*C_ + t];
  if (t < 3) lc_xyz[(b*G_ + g)*3 + t] = xyz[row*3 + t];
}

// ---------------------------------------------------------------------------
// Cross-half KNN + neighbor gathers. One block per (b, 16-query tile).
// Dot products via v_wmma_f32_16x16x4_f32 (K=4: xyz padded with 0).
// ---------------------------------------------------------------------------
__global__ __launch_bounds__(256) void knn_kernel(const float* __restrict__ xyz,
                                                  const float* __restrict__ feat,
                                                  const float* __restrict__ lc_xyz,
                                                  float* __restrict__ knn_xyz,
                                                  float* __restrict__ knn_feat) {
  extern __shared__ float smem[];
  float* sdist = smem;               // 16 * 4096 floats (256KB)
  float* srsq  = smem + 16*HALF_;    // 4096 floats (16KB)
  __shared__ float sqsq[16];
  __shared__ int   snn[16*K_];

  const int tid  = threadIdx.x;
  const int lane = tid & 31;
  const int wave = tid >> 5;
  const int b    = blockIdx.x >> 6;
  const int qt   = blockIdx.x & 63;
  const int g0   = qt * 16;
  const int ref_base = (g0 < 512) ? HALF_ : 0;   // right queries search left half & v.v.
  const int rrow0 = b*N_ + ref_base;

  // ref squared norms
  for (int r = tid; r < HALF_; r += 256) {
    const float x = xyz[(rrow0+r)*3], y = xyz[(rrow0+r)*3+1], z = xyz[(rrow0+r)*3+2];
    srsq[r] = x*x + y*y + z*z;
  }
  // query squared norms
  if (tid < 16) {
    const float* q = lc_xyz + (b*G_ + g0 + tid)*3;
    sqsq[tid] = q[0]*q[0] + q[1]*q[1] + q[2]*q[2];
  }
  __syncthreads();

  // Operand layout for 16x4 f32 WMMA: lane&15 = M (or N), (lane>>4)*2 = K slice.
  const int  m   = lane & 15;
  const bool kb0 = (lane < 16);      // K slice {0,1} for lanes 0-15, {2,3} for 16-31
  {
    const float* q = lc_xyz + (b*G_ + g0 + m)*3;
    const float q0 = q[0], q1 = q[1], q2 = q[2];  // unconditional: no exec-branch
    v2f av;
    av.x = kb0 ? q0 : q2;
    av.y = kb0 ? q1 : 0.0f;

    // 8 waves x 32 tiles cover 4096 refs
    for (int t = 0; t < 32; ++t) {
      const int tile = wave*32 + t;
      const int rr   = tile*16 + m;
      const float* rp = xyz + (rrow0 + rr)*3;
      const float r0 = rp[0], r1 = rp[1], r2 = rp[2];
      v2f bbv;
      bbv.x = kb0 ? r0 : r2;
      bbv.y = kb0 ? r1 : 0.0f;
      v8f acc = {0.f,0.f,0.f,0.f,0.f,0.f,0.f,0.f};
      v8f dot = __builtin_amdgcn_wmma_f32_16x16x4_f32(
          /*neg_a=*/false, av, /*neg_b=*/false, bbv,
          /*c_mod=*/(short)0, acc, /*reuse_a=*/false, /*reuse_b=*/false);
      const float rs = srsq[rr];
      #pragma unroll
      for (int j = 0; j < 8; ++j) {
        const int M = j + ((lane >> 4) << 3);     // D layout: lanes16-31 hold M+8
        const float dd = fmaxf(sqsq[M] + rs - 2.0f * dot[j], 0.0f);
        sdist[M*HALF_ + rr] = dd;
      }
    }
  }
  __syncthreads();

  // ---- top-32 selection: 2 queries per wave ------------------------------
  // Each lane caches its local top-2 of its 128 columns; extraction is pure
  // shfl butterflies; a lane re-scans only when both cached entries are used.
  for (int s = 0; s < 2; ++s) {
    const int qi = wave*2 + s;
    const float* dq = sdist + qi*HALF_;
    unsigned long long msk0 = 0ull, msk1 = 0ull;   // taken-mask, bit i -> col i*32+lane

    float v1 = 1e30f, v2 = 1e30f;
    int   i1 = 0x7fffffff, i2 = 0x7fffffff;
    for (int i = 0; i < 128; ++i) {
      const int col = i*32 + lane;
      const float v = dq[col];
      if (v < v1 || (v == v1 && col < i1)) { v2 = v1; i2 = i1; v1 = v; i1 = col; }
      else if (v < v2 || (v == v2 && col < i2)) { v2 = v; i2 = col; }
    }
    bool have2 = true;

    for (int k = 0; k < K_; ++k) {
      float bv = v1; int bi = i1;
      #pragma unroll
      for (int off = 16; off > 0; off >>= 1) {
        const float ov = __shfl_xor(bv, off, 32);
        const int   oi = __shfl_xor(bi, off, 32);
        if (ov < bv || (ov == bv && oi < bi)) { bv = ov; bi = oi; }
      }
      if (lane == 0) snn[qi*K_ + k] = bi;

      if ((bi & 31) == lane) {                 // this lane owns the winner
        const int ii = bi >> 5;
        if (ii < 64) msk0 |= 1ull << ii; else msk1 |= 1ull << (ii - 64);
        if (have2) {
          v1 = v2; i1 = i2;
          v2 = 1e30f; i2 = 0x7fffffff;
          have2 = false;
        } else {                               // rare: rebuild local top-2
          v1 = 1e30f; i1 = 0x7fffffff;
          v2 = 1e30f; i2 = 0x7fffffff;
          for (int i = 0; i < 128; ++i) {
            const bool taken = (((i < 64) ? (msk0 >> i) : (msk1 >> (i - 64))) & 1ull) != 0ull;
            if (taken) continue;
            const int col = i*32 + lane;
            const float v = dq[col];
            if (v < v1 || (v == v1 && col < i1)) { v2 = v1; i2 = i1; v1 = v; i1 = col; }
            else if (v < v2 || (v == v2 && col < i2)) { v2 = v; i2 = col; }
          }
          have2 = true;
        }
      }
    }
  }
  __syncthreads();

  // neighbor gathers: 512 (q,k) pairs, full-wave coalesced channel writes
  for (int pair = wave; pair < 16*K_; pair += 8) {
    const int qi  = pair >> 5;
    const int k   = pair & 31;
    const int idx = snn[qi*K_ + k];
    const int row = rrow0 + idx;
    const int g   = g0 + qi;
    const int obase = (b*G_ + g)*K_ + k;
    knn_feat[obase*67 + lane]      = feat[row*C_ + lane];
    knn_feat[obase*67 + 32 + lane] = feat[row*C_ + 32 + lane];
    if (lane < 3) {
      const float c = xyz[row*3 + lane];
      knn_feat[obase*67 + 64 + lane] = c;
      knn_xyz[obase*3 + lane]        = c;
    }
  }
}

// ---------------------------------------------------------------------------
extern "C" void kernel_launch(void* const* d_in, const int* in_sizes, int n_in,
                              void* d_out, int out_size, void* d_ws, size_t ws_size,
                              hipStream_t stream) {
  const float* xyz  = (const float*)d_in[0];
  const float* feat = (const float*)d_in[1];

  float* lc_xyz   = (float*)d_out;
  float* lc_feat  = lc_xyz  + (size_t)B_*G_*3;
  float* knn_xyz  = lc_feat + (size_t)B_*G_*C_;
  float* knn_feat = knn_xyz + (size_t)B_*G_*K_*3;

  int* sel = (int*)d_ws;   // 32 * 512 ints

  const size_t fps_lds = (size_t)HALF_ * 3 * sizeof(float);
  fps_kernel<<<B_*2, 256, fps_lds, stream>>>(xyz, sel);

  lc_gather_kernel<<<B_*G_, 64, 0, stream>>>(xyz, feat, sel, lc_xyz, lc_feat);

  const size_t knn_lds = (size_t)(16*HALF_ + HALF_) * sizeof(float);  // 272KB (<320KB WGP LDS)
  (void)hipFuncSetAttribute((const void*)knn_kernel,
                            hipFuncAttributeMaxDynamicSharedMemorySize, (int)knn_lds);
  knn_kernel<<<B_*64, 256, knn_lds, stream>>>(xyz, feat, lc_xyz, knn_xyz, knn_feat);
}